// EdgeConvLayer_18236431139304
// MI455X (gfx1250) — compile-verified
//
#include <hip/hip_runtime.h>
#include <hip/hip_bf16.h>

typedef float v2f __attribute__((ext_vector_type(2)));
typedef float v8f __attribute__((ext_vector_type(8)));

// Pair-interleaved LDS layout: wc[p][o'][2] holds Wc[2p][o'], Wc[2p+1][o'].
// Row stride 288 floats (= 128*2 + 32 pad) so p -> p+1 shifts 32 banks:
// lanes 0-15 (p) and lanes 16-31 (p+1) hit disjoint bank halves on b64 reads.
#define PAIR_STRIDE 288

// ---------------------------------------------------------------------------
// Kernel 1: H[node][0:64]   = X @ (W1 - W2)          (per-node "center" term)
//           H[node][64:128] = X @ W2                 (per-node "neighbor" term)
// X is x[B,C,N] viewed as [32768 nodes x 64 feats] (feature stride = N floats).
// One wave computes a 16-node x 128-col tile via v_wmma_f32_16x16x4_f32.
// ---------------------------------------------------------------------------
__global__ __launch_bounds__(256) void edgeconv_gemm(const float* __restrict__ x,
                                                     const float* __restrict__ W,
                                                     float* __restrict__ H) {
    __shared__ float wc[32 * PAIR_STRIDE];     // 36.9 KB of 320 KB/WGP

    const int tid = threadIdx.x;
    // Build combined weight Wc[c][o']: o'<64 -> W1[c][o']-W2[c][o'], o'>=64 -> W2[c][o'-64]
    for (int i = tid; i < 64 * 128; i += 256) {
        int c = i >> 7;
        int o = i & 127;
        float v;
        if (o < 64) v = W[c * 64 + o] - W[(64 + c) * 64 + o];
        else        v = W[(64 + c) * 64 + (o - 64)];
        wc[(c >> 1) * PAIR_STRIDE + o * 2 + (c & 1)] = v;
    }
    __syncthreads();

    const int lane = tid & 31;
    const int wv   = tid >> 5;
    const int tile = blockIdx.x * 8 + wv;      // 2048 tiles of 16 nodes
    const int m    = lane & 15;                // M index within tile
    const int kh   = lane >> 4;                // K-half select (ISA f32 A layout)
    const int node = tile * 16 + m;
    const int bb   = node >> 12;               // batch (N = 4096)
    const int nn   = node & 4095;
    const float* xb = x + ((size_t)bb * 64) * 4096 + nn;

    // A-fragments: 16 k-steps of K=4. Lane l holds A[m][c0], A[m][c0+1], c0 = 4*kk+2*kh.
    v2f a[16];
#pragma unroll
    for (int kk = 0; kk < 16; ++kk) {
        int c0 = 4 * kk + 2 * kh;
        a[kk].x = xb[(size_t)c0 * 4096];
        a[kk].y = xb[(size_t)(c0 + 1) * 4096];
    }

#pragma unroll
    for (int ot = 0; ot < 8; ++ot) {           // 8 column tiles of 16
        v8f acc = {};
#pragma unroll
        for (int kk = 0; kk < 16; ++kk) {
            // pair index p = (4*kk + 2*kh)/2 = 2*kk + kh ; one aligned ds_load_b64
            const v2f bf = *(const v2f*)&wc[(2 * kk + kh) * PAIR_STRIDE + (ot * 16 + m) * 2];
            acc = __builtin_amdgcn_wmma_f32_16x16x4_f32(
                false, a[kk], false, bf, (short)0, acc, false, false);
        }
        // D layout: VGPR j, lane l -> D[j + 8*(l>>4)][(l&15) + 16*ot]
        float* Hp = H + ((size_t)tile * 16 + 8 * kh) * 128 + ot * 16 + m;
#pragma unroll
        for (int j = 0; j < 8; ++j)
            Hp[(size_t)j * 128] = acc[j];
    }
}

// ---------------------------------------------------------------------------
// Kernel 2: out[b,o,n] = relu( H[node][o] + bias[o] + max_k H[b*4096+src_k][64+o] )
// One wave per node; lane covers 2 output columns (float2). Gather rows are
// 256B coalesced and L2-resident (H = 16 MB << 192 MB L2).
// ---------------------------------------------------------------------------
__global__ __launch_bounds__(256) void edgeconv_aggregate(const float* __restrict__ H,
                                                          const int* __restrict__ src,
                                                          const float* __restrict__ bias,
                                                          float* __restrict__ out) {
    const int lane = threadIdx.x & 31;
    const int wv   = threadIdx.x >> 5;
    const int node = blockIdx.x * 8 + wv;      // 32768 nodes total
    const int bb   = node >> 12;
    const int nn   = node & 4095;

    // lanes 0-15 fetch the 16 neighbor indices (64B coalesced); broadcast via shfl
    int s = src[node * 16 + (lane & 15)];

    float m0 = -3.402823466e38f, m1 = -3.402823466e38f;
#pragma unroll
    for (int k = 0; k < 16; ++k) {
        int sk = __shfl(s, k, 32);
        const v2f* crow = (const v2f*)(H + ((size_t)(bb * 4096 + sk)) * 128 + 64);
        v2f c = crow[lane];
        m0 = fmaxf(m0, c.x);
        m1 = fmaxf(m1, c.y);
    }

    v2f aA = ((const v2f*)(H + (size_t)node * 128))[lane];
    v2f bV = ((const v2f*)bias)[lane];
    float r0 = fmaxf(aA.x + bV.x + m0, 0.0f);
    float r1 = fmaxf(aA.y + bV.y + m1, 0.0f);

    size_t obase = ((size_t)bb * 64 + 2 * lane) * 4096 + nn;
    out[obase]        = r0;
    out[obase + 4096] = r1;
}

extern "C" void kernel_launch(void* const* d_in, const int* in_sizes, int n_in,
                              void* d_out, int out_size, void* d_ws, size_t ws_size,
                              hipStream_t stream) {
    const float* x  = (const float*)d_in[0];   // [8,64,4096,1] f32
    const int*   ei = (const int*)d_in[1];     // [2,8,4096,16] i32; src half at offset 0
    const float* W  = (const float*)d_in[2];   // [128,64] f32
    const float* b  = (const float*)d_in[3];   // [64] f32
    float* out = (float*)d_out;                // [8,64,4096,1] f32
    float* H   = (float*)d_ws;                 // 32768 x 128 f32 = 16 MB scratch

    edgeconv_gemm<<<256, 256, 0, stream>>>(x, W, H);
    edgeconv_aggregate<<<4096, 256, 0, stream>>>(H, ei, b, out);
}